// SAGE_74148315398467
// MI455X (gfx1250) — compile-verified
//
#include <hip/hip_runtime.h>
#include <hip/hip_bf16.h>

// ---------------- problem constants (from reference) ----------------
#define N1_SRC 1171456
#define N1_TGT 45056
#define N2_TGT 4096
#define E1 1126400
#define E2 40960
#define IN_C 128
#define HID_C 256
#define OUT_C 64

// ---------------- workspace layout (floats) ----------------
// [agg1 | cnt1 | agg2 | cnt2 | h]  -- first four are zero-initialized each call
#define AGG1_OFF 0ull
#define AGG1_LEN ((size_t)N1_TGT * IN_C)            // 5,767,168
#define CNT1_OFF (AGG1_OFF + AGG1_LEN)
#define CNT1_LEN ((size_t)N1_TGT)                   // 45,056
#define AGG2_OFF (CNT1_OFF + CNT1_LEN)
#define AGG2_LEN ((size_t)N2_TGT * HID_C)           // 1,048,576
#define CNT2_OFF (AGG2_OFF + AGG2_LEN)
#define CNT2_LEN ((size_t)N2_TGT)                   // 4,096
#define H_OFF    (CNT2_OFF + CNT2_LEN)              // 6,864,896
#define H_LEN    ((size_t)N1_TGT * HID_C)           // 11,534,336
#define ZERO_SPAN (H_OFF)                           // everything before h

typedef __attribute__((ext_vector_type(2))) float v2f;
typedef __attribute__((ext_vector_type(8))) float v8f;

// ---------------- WMMA helper: D = A(16x4,f32) * B(4x16,f32) + C ----------------
__device__ __forceinline__ v8f wmma_f32_k4(v2f a, v2f b, v8f c) {
  return __builtin_amdgcn_wmma_f32_16x16x4_f32(
      /*neg_a=*/false, a, /*neg_b=*/false, b,
      /*c_mod=*/(short)0, c, /*reuse_a=*/false, /*reuse_b=*/false);
}

// ---------------- zero-fill (float4 granularity) ----------------
__global__ __launch_bounds__(256) void zero_f4_kernel(float4* __restrict__ p, int n4) {
  int i = blockIdx.x * 256 + threadIdx.x;
  if (i < n4) p[i] = make_float4(0.f, 0.f, 0.f, 0.f);
}

// ---------------- edge scatter-add: agg[dst] += xsrc[src]; cnt[dst] += 1 -------
// (1<<SHIFT) lanes per edge, float4 per lane; CHAN = channels.
template <int CHAN, int SHIFT>
__global__ __launch_bounds__(256) void scatter_add_kernel(
    const float* __restrict__ xsrc, const int* __restrict__ src,
    const int* __restrict__ dst, float* __restrict__ agg,
    float* __restrict__ cnt, int nEdges)
{
  unsigned long long gid = (unsigned long long)blockIdx.x * 256ull + threadIdx.x;
  int e = (int)(gid >> SHIFT);
  int q = (int)(gid & ((1u << SHIFT) - 1u));
  if (e >= nEdges) return;
  int s = src[e];
  int d = dst[e];
  const float4 v = *((const float4*)(xsrc + (size_t)s * CHAN) + q);
  float* o = agg + (size_t)d * CHAN + (size_t)q * 4;
  atomicAdd(o + 0, v.x);
  atomicAdd(o + 1, v.y);
  atomicAdd(o + 2, v.z);
  atomicAdd(o + 3, v.w);
  if (q == 0) atomicAdd(cnt + d, 1.0f);
}

// ---------------- fused SAGE conv via WMMA (all shapes compile-time) ----------
// out[m,n] = (agg[m,:]/max(cnt[m],1)) @ Wl + xr[m,:] @ Wr + bias[n]  (opt. ReLU)
// One wave -> 16x64 output tile. Requires M%16==0, N%64==0, K%4==0 (all hold).
// NGROUPS = N/64 must be a power of two (4 for layer 1, 1 for layer 2).
template <int K, int N, int NGROUPS, int RELU>
__global__ __launch_bounds__(256) void sage_conv_wmma_kernel(
    const float* __restrict__ agg, const float* __restrict__ cnt,
    const float* __restrict__ xr,  const float* __restrict__ Wl,
    const float* __restrict__ Wr,  const float* __restrict__ bias,
    float* __restrict__ out)
{
  static_assert((NGROUPS & (NGROUPS - 1)) == 0, "NGROUPS must be pow2");
  const int lane   = threadIdx.x & 31;
  const int waveId = (int)((blockIdx.x * 256u + threadIdx.x) >> 5);
  const int mtile  = waveId / NGROUPS;          // shift (pow2)
  const int ngrp   = waveId & (NGROUPS - 1);    // mask
  const int m0 = mtile * 16;
  const int n0 = ngrp * 64;
  const int r  = lane & 15;       // A: row M / B,D: column N
  const int kh = lane >> 4;       // half-wave selects K pair (A) / B K-row pair

  const float inv = 1.0f / fmaxf(cnt[m0 + r], 1.0f);

  v8f acc0 = {}; v8f acc1 = {}; v8f acc2 = {}; v8f acc3 = {};

  const float* Ar = agg + (size_t)(m0 + r) * K + kh * 2;   // aggregated (pre-mean)
  const float* Xr = xr  + (size_t)(m0 + r) * K + kh * 2;   // root features
  const float* WlB = Wl + (size_t)(kh * 2) * N + n0 + r;
  const float* WrB = Wr + (size_t)(kh * 2) * N + n0 + r;

  // GEMM 1: mean(agg) @ Wl   (mean folded into A-fragment load)
#pragma unroll 4
  for (int k0 = 0; k0 < K; k0 += 4) {
    v2f a; a.x = Ar[k0] * inv; a.y = Ar[k0 + 1] * inv;
    const float* B0 = WlB + (size_t)k0 * N;
    const float* B1 = B0 + N;
    v2f b0; b0.x = B0[0];  b0.y = B1[0];
    v2f b1; b1.x = B0[16]; b1.y = B1[16];
    v2f b2; b2.x = B0[32]; b2.y = B1[32];
    v2f b3; b3.x = B0[48]; b3.y = B1[48];
    acc0 = wmma_f32_k4(a, b0, acc0);
    acc1 = wmma_f32_k4(a, b1, acc1);
    acc2 = wmma_f32_k4(a, b2, acc2);
    acc3 = wmma_f32_k4(a, b3, acc3);
  }
  // GEMM 2: x_tgt @ Wr  (accumulate into same tile)
#pragma unroll 4
  for (int k0 = 0; k0 < K; k0 += 4) {
    v2f a; a.x = Xr[k0]; a.y = Xr[k0 + 1];
    const float* B0 = WrB + (size_t)k0 * N;
    const float* B1 = B0 + N;
    v2f b0; b0.x = B0[0];  b0.y = B1[0];
    v2f b1; b1.x = B0[16]; b1.y = B1[16];
    v2f b2; b2.x = B0[32]; b2.y = B1[32];
    v2f b3; b3.x = B0[48]; b3.y = B1[48];
    acc0 = wmma_f32_k4(a, b0, acc0);
    acc1 = wmma_f32_k4(a, b1, acc1);
    acc2 = wmma_f32_k4(a, b2, acc2);
    acc3 = wmma_f32_k4(a, b3, acc3);
  }

  // epilogue: +bias, optional ReLU, store. D layout: VGPR v -> row v + 8*kh.
  const float bb0 = bias[n0 + r];
  const float bb1 = bias[n0 + 16 + r];
  const float bb2 = bias[n0 + 32 + r];
  const float bb3 = bias[n0 + 48 + r];
#pragma unroll
  for (int v = 0; v < 8; ++v) {
    const int row = m0 + kh * 8 + v;
    float* orow = out + (size_t)row * N + n0 + r;
    float o0 = acc0[v] + bb0;
    float o1 = acc1[v] + bb1;
    float o2 = acc2[v] + bb2;
    float o3 = acc3[v] + bb3;
    if (RELU) {
      o0 = fmaxf(o0, 0.f); o1 = fmaxf(o1, 0.f);
      o2 = fmaxf(o2, 0.f); o3 = fmaxf(o3, 0.f);
    }
    orow[0]  = o0;
    orow[16] = o1;
    orow[32] = o2;
    orow[48] = o3;
  }
}

// ---------------- log_softmax over 64 classes, one wave32 per row -------------
__global__ __launch_bounds__(256) void log_softmax64_kernel(float* __restrict__ out, int rows)
{
  const int lane = threadIdx.x & 31;
  const int row  = (int)((blockIdx.x * 256u + threadIdx.x) >> 5);
  if (row >= rows) return;
  float* p = out + (size_t)row * 64;
  float v0 = p[lane];
  float v1 = p[lane + 32];
  float m = fmaxf(v0, v1);
#pragma unroll
  for (int off = 16; off >= 1; off >>= 1)
    m = fmaxf(m, __shfl_xor(m, off, 32));
  float s = expf(v0 - m) + expf(v1 - m);
#pragma unroll
  for (int off = 16; off >= 1; off >>= 1)
    s += __shfl_xor(s, off, 32);
  const float ls = logf(s);
  p[lane]      = v0 - m - ls;
  p[lane + 32] = v1 - m - ls;
}

// ---------------- launcher ----------------
extern "C" void kernel_launch(void* const* d_in, const int* in_sizes, int n_in,
                              void* d_out, int out_size, void* d_ws, size_t ws_size,
                              hipStream_t stream)
{
  const float* x    = (const float*)d_in[0];
  const float* Wl1  = (const float*)d_in[1];
  const float* Wr1  = (const float*)d_in[2];
  const float* b1   = (const float*)d_in[3];
  const float* Wl2  = (const float*)d_in[4];
  const float* Wr2  = (const float*)d_in[5];
  const float* b2   = (const float*)d_in[6];
  const int*   src1 = (const int*)d_in[7];
  const int*   dst1 = (const int*)d_in[8];
  const int*   src2 = (const int*)d_in[9];
  const int*   dst2 = (const int*)d_in[10];

  float* ws   = (float*)d_ws;
  float* agg1 = ws + AGG1_OFF;
  float* cnt1 = ws + CNT1_OFF;
  float* agg2 = ws + AGG2_OFF;
  float* cnt2 = ws + CNT2_OFF;
  float* h    = ws + H_OFF;
  float* outp = (float*)d_out;

  // 1) zero the accumulators (agg1|cnt1|agg2|cnt2 are contiguous)
  {
    int n4 = (int)(ZERO_SPAN / 4);                 // 1,716,224 float4
    int blocks = (n4 + 255) / 256;
    zero_f4_kernel<<<blocks, 256, 0, stream>>>((float4*)ws, n4);
  }

  // 2) layer-1 neighbor aggregation: 32 lanes/edge, float4 each (128 ch)
  {
    unsigned long long threads = (unsigned long long)E1 * 32ull;
    int blocks = (int)(threads / 256ull);          // 140,800 exactly
    scatter_add_kernel<IN_C, 5><<<blocks, 256, 0, stream>>>(
        x, src1, dst1, agg1, cnt1, E1);
  }

  // 3) layer 1: h = relu(mean(agg1) @ Wl1 + x[:N1_TGT] @ Wr1 + b1)
  {
    int waves  = (N1_TGT / 16) * (HID_C / 64);     // 2816 * 4 = 11264
    int blocks = waves / 8;                        // 1408
    sage_conv_wmma_kernel<IN_C, HID_C, HID_C / 64, 1><<<blocks, 256, 0, stream>>>(
        agg1, cnt1, x, Wl1, Wr1, b1, h);
  }

  // 4) layer-2 neighbor aggregation: 64 lanes/edge, float4 each (256 ch)
  {
    unsigned long long threads = (unsigned long long)E2 * 64ull;
    int blocks = (int)(threads / 256ull);          // 10,240 exactly
    scatter_add_kernel<HID_C, 6><<<blocks, 256, 0, stream>>>(
        h, src2, dst2, agg2, cnt2, E2);
  }

  // 5) layer 2: logits = mean(agg2) @ Wl2 + h[:N2_TGT] @ Wr2 + b2  -> d_out
  {
    int waves  = (N2_TGT / 16) * (OUT_C / 64);     // 256
    int blocks = waves / 8;                        // 32
    sage_conv_wmma_kernel<HID_C, OUT_C, OUT_C / 64, 0><<<blocks, 256, 0, stream>>>(
        agg2, cnt2, h, Wl2, Wr2, b2, outp);
  }

  // 6) log_softmax in place on d_out (recomputed fully each call; deterministic)
  {
    int blocks = N2_TGT / 8;                       // 512
    log_softmax64_kernel<<<blocks, 256, 0, stream>>>(outp, N2_TGT);
  }
}